// ConvDeepSet_28647431864437
// MI455X (gfx1250) — compile-verified
//
#include <hip/hip_runtime.h>
#include <hip/hip_bf16.h>

// Problem constants (fixed by the reference).
#define BATCH   16
#define N_IN    512
#define N_OUT   4096
#define OUT_CH  64

typedef __attribute__((ext_vector_type(2))) float v2f;
typedef __attribute__((ext_vector_type(8))) float v8f;

// log2(e): pre-scale exponent constants so the inner loop uses raw v_exp_f32.
#define LOG2E 1.4426950408889634f

// ---------------------------------------------------------------------------
// Kernel 1: RBF aggregation. One thread per (b, m) output position.
// Each 256-thread block covers 256 consecutive m within a single batch b
// (N_OUT / 256 = 16 blocks per batch), so x/y of that batch are staged in LDS.
// Inner loop: d2 = (x-t)^2, then two raw v_exp_f32 (base-2, constants
// pre-scaled by log2e) -> transcendental-rate bound, other VALU co-executes.
// Writes feats = [dens, conv/(dens+eps)] as float2 into workspace.
// ---------------------------------------------------------------------------
__global__ __launch_bounds__(256) void convdeepset_agg(
    const float* __restrict__ x, const float* __restrict__ y,
    const float* __restrict__ t, const float* __restrict__ sigma,
    float* __restrict__ feats /* [BATCH*N_OUT][2] */)
{
    __shared__ float xs[N_IN];
    __shared__ float ys[N_IN];

    const int p = blockIdx.x * 256 + threadIdx.x;   // 0 .. BATCH*N_OUT-1
    const int b = p >> 12;                          // p / N_OUT (uniform per block)

    // Cooperative stage of this batch's points into LDS (4 KB).
    for (int i = threadIdx.x; i < N_IN; i += 256) {
        xs[i] = x[b * N_IN + i];
        ys[i] = y[b * N_IN + i];
    }
    __syncthreads();

    const float s0 = __expf(sigma[0]);
    const float s1 = __expf(sigma[1]);
    const float k0 = (-0.5f * LOG2E) / (s0 * s0);   // exp(-0.5 d2/s0^2) = exp2(k0*d2)
    const float k1 = (-0.5f * LOG2E) / (s1 * s1);

    const float tm = t[p];

    float dens = 0.0f;
    float conv = 0.0f;
#pragma unroll 4
    for (int n = 0; n < N_IN; ++n) {
        const float d  = xs[n] - tm;
        const float d2 = d * d;
        dens += __builtin_amdgcn_exp2f(k0 * d2);
        conv += ys[n] * __builtin_amdgcn_exp2f(k1 * d2);
    }

    feats[2 * p + 0] = dens;
    feats[2 * p + 1] = conv / (dens + 1e-8f);
}

// ---------------------------------------------------------------------------
// Kernel 2: projection out = feats @ W^T + bias via V_WMMA_F32_16X16X4_F32.
// feats is (65536 x 2) -> K padded to 4 with zeros; W is (64 x 2) row-major.
// One wave per 16-row tile of feats; 4 WMMAs cover the 64 output channels.
//
// A fragment (16x4 f32, 2 VGPRs): lanes 0-15 hold M=0..15 with VGPR0=K0,
// VGPR1=K1; lanes 16-31 hold K=2,3 -> forced to zero (padding).
// B fragment (4x16 f32, 2 VGPRs): lanes 0-15 hold N=0..15 with VGPR0=K0 row,
// VGPR1=K1 row; lanes 16-31 hold K=2,3 rows -> zero.
// C/D (8 VGPRs): VGPR r, lanes 0-15 -> D[M=r][N=lane]; lanes 16-31 ->
// D[M=r+8][N=lane-16]. Bias depends only on N -> preload C with bias[col]
// so the WMMA accumulator performs the bias add (no post-WMMA VALU adds).
// ---------------------------------------------------------------------------
__global__ __launch_bounds__(128) void convdeepset_proj(
    const float* __restrict__ feats, const float* __restrict__ W,
    const float* __restrict__ bias, float* __restrict__ out)
{
    const int lane = threadIdx.x & 31;
    const int wave = threadIdx.x >> 5;
    const int tile = blockIdx.x * 4 + wave;     // 16-row tile index
    const int row0 = tile * 16;
    const int l    = lane & 15;
    const bool lo  = lane < 16;                 // lanes 16-31 carry K=2,3 padding

    // A fragment: always load a valid address, then mask (keeps flow convergent).
    v2f a;
    {
        const float a0 = feats[(row0 + l) * 2 + 0];
        const float a1 = feats[(row0 + l) * 2 + 1];
        a.x = lo ? a0 : 0.0f;
        a.y = lo ? a1 : 0.0f;
    }

#pragma unroll
    for (int j = 0; j < 4; ++j) {
        const int col = j * 16 + l;             // output channel this lane owns
        // B fragment for this 16-column tile of W^T.
        v2f bf;
        {
            const float b0 = W[col * 2 + 0];
            const float b1 = W[col * 2 + 1];
            bf.x = lo ? b0 : 0.0f;
            bf.y = lo ? b1 : 0.0f;
        }

        // C preloaded with the bias for this lane's column (same for all rows).
        const float bv = bias[col];
        v8f c = {bv, bv, bv, bv, bv, bv, bv, bv};

        // (neg_a, A, neg_b, B, c_mod, C, reuse_a, reuse_b)
        c = __builtin_amdgcn_wmma_f32_16x16x4_f32(
                false, a, false, bf, (short)0, c, false, false);

        const int r0 = row0 + (lo ? 0 : 8);
#pragma unroll
        for (int r = 0; r < 8; ++r) {
            out[(r0 + r) * OUT_CH + col] = c[r];
        }
    }
}

extern "C" void kernel_launch(void* const* d_in, const int* in_sizes, int n_in,
                              void* d_out, int out_size, void* d_ws, size_t ws_size,
                              hipStream_t stream) {
    const float* x     = (const float*)d_in[0];   // (16, 512, 1)
    const float* y     = (const float*)d_in[1];   // (16, 512, 1)
    const float* t     = (const float*)d_in[2];   // (16, 4096, 1)
    const float* sigma = (const float*)d_in[3];   // (2,)
    const float* W     = (const float*)d_in[4];   // (64, 2)
    const float* bias  = (const float*)d_in[5];   // (64,)
    float* out   = (float*)d_out;                 // (16, 4096, 64)
    float* feats = (float*)d_ws;                  // (16*4096, 2) = 512 KB

    // Kernel 1: 65536 positions / 256 threads = 256 blocks.
    convdeepset_agg<<<(BATCH * N_OUT) / 256, 256, 0, stream>>>(x, y, t, sigma, feats);

    // Kernel 2: 4096 M-tiles, 4 waves (tiles) per 128-thread block = 1024 blocks.
    convdeepset_proj<<<(BATCH * N_OUT / 16) / 4, 128, 0, stream>>>(feats, W, bias, out);
}